// ViewLearner_300647711305
// MI455X (gfx1250) — compile-verified
//
#include <hip/hip_runtime.h>

typedef float v2f __attribute__((ext_vector_type(2)));
typedef float v8f __attribute__((ext_vector_type(8)));

#define HIDDEN   128     // per-node embedding width
#define MLP      64      // hidden units of edge MLP
#define PITCH    260     // LDS row pitch for W1^T: bank = (4m + k) % 64 -> conflict-free b64 loads
#define THREADS  256     // 8 waves (wave32)
#define EDGES_PER_BLOCK 256   // 8 waves * 2 tiles * 16 edges

// D[m][e] = sum_k W1[k][m] * edge[e][k];  edge[e][k] = k<128 ? emb[src[e]][k] : emb[dst[e]][k-128]
__global__ __launch_bounds__(THREADS)
void edge_mlp_wmma(const float* __restrict__ node_emb,
                   const long long* __restrict__ edge_index,  // [2, E]
                   const float* __restrict__ W1,              // [256, 64] row-major
                   const float* __restrict__ b1,              // [64]
                   const float* __restrict__ W2,              // [64]
                   const float* __restrict__ b2,              // [1]
                   float* __restrict__ out,                   // [E]
                   int n_edges)
{
    extern __shared__ float smem[];
    float* w1t = smem;                 // [64][PITCH]  w1t[m][k] = W1[k*64+m]
    float* b1s = smem + MLP * PITCH;   // [64]
    float* w2s = b1s + MLP;            // [64]

    const int tid = threadIdx.x;
    // Stage W1^T into padded LDS (coalesced global reads, one-time 64KB)
    for (int i = tid; i < 2 * HIDDEN * MLP; i += THREADS) {
        const int k = i >> 6;          // 0..255
        const int m = i & 63;          // 0..63
        w1t[m * PITCH + k] = W1[i];    // W1[i] == W1[k*64 + m]
    }
    if (tid < MLP) { b1s[tid] = b1[tid]; w2s[tid] = W2[tid]; }
    __syncthreads();

    const int wave = tid >> 5;
    const int lane = tid & 31;
    const int m16  = lane & 15;        // M/N position within a 16x16 tile
    const int kh   = lane >> 4;        // half-wave selects K sub-pair (ISA 16x16x4 layout)

    const int ebase = (blockIdx.x * 8 + wave) * 32;   // 32 edges per wave
    const int e0 = ebase + m16;        // edge for B-tile 0 (this lane's column)
    const int e1 = e0 + 16;            // edge for B-tile 1

    const long long* srcI = edge_index;            // first E entries
    const long long* dstI = edge_index + n_edges;  // second E entries

    const float* s0 = node_emb + (size_t)srcI[e0] * HIDDEN;
    const float* d0 = node_emb + (size_t)dstI[e0] * HIDDEN;
    const float* s1 = node_emb + (size_t)srcI[e1] * HIDDEN;
    const float* d1 = node_emb + (size_t)dstI[e1] * HIDDEN;

    // Get the dst rows moving toward L0/L2 while the src half is consumed.
    __builtin_prefetch(d0, 0, 0);
    __builtin_prefetch(d0 + 64, 0, 0);
    __builtin_prefetch(d1, 0, 0);
    __builtin_prefetch(d1 + 64, 0, 0);

    v8f acc0[4] = {};   // D tiles, hidden rows 16t..16t+15, edges ebase..+15
    v8f acc1[4] = {};   // same rows, edges ebase+16..+31

    const int cofs = 2 * kh;           // lane-group K offset within a K-step of 4

    // First half of K: source-node features (weight cols 0..127)
#pragma unroll 4
    for (int k0 = 0; k0 < HIDDEN; k0 += 4) {
        const int c = k0 + cofs;       // even -> 8B aligned float2
        const v2f bv0 = *(const v2f*)(s0 + c);
        const v2f bv1 = *(const v2f*)(s1 + c);
#pragma unroll
        for (int t = 0; t < 4; ++t) {
            const v2f av = *(const v2f*)(&w1t[(t * 16 + m16) * PITCH + c]);
            acc0[t] = __builtin_amdgcn_wmma_f32_16x16x4_f32(
                false, av, false, bv0, (short)0, acc0[t], false, false);
            acc1[t] = __builtin_amdgcn_wmma_f32_16x16x4_f32(
                false, av, false, bv1, (short)0, acc1[t], false, false);
        }
    }
    // Second half of K: destination-node features (weight cols 128..255)
#pragma unroll 4
    for (int k0 = 0; k0 < HIDDEN; k0 += 4) {
        const int c = k0 + cofs;
        const v2f bv0 = *(const v2f*)(d0 + c);
        const v2f bv1 = *(const v2f*)(d1 + c);
#pragma unroll
        for (int t = 0; t < 4; ++t) {
            const v2f av = *(const v2f*)(&w1t[(t * 16 + m16) * PITCH + (HIDDEN + c)]);
            acc0[t] = __builtin_amdgcn_wmma_f32_16x16x4_f32(
                false, av, false, bv0, (short)0, acc0[t], false, false);
            acc1[t] = __builtin_amdgcn_wmma_f32_16x16x4_f32(
                false, av, false, bv1, (short)0, acc1[t], false, false);
        }
    }

    // Epilogue: out[e] = sum_m relu(h[m][e] + b1[m]) * W2[m] + b2
    // Lane holds D rows m = 16t + r + 8*kh for its edge column; lane pair (l, l+16)
    // together covers all 64 hidden units -> one shfl_xor(16) completes the dot.
    float p0 = 0.f, p1 = 0.f;
#pragma unroll
    for (int t = 0; t < 4; ++t) {
#pragma unroll
        for (int r = 0; r < 8; ++r) {
            const int m = t * 16 + r + 8 * kh;
            const float bb = b1s[m];
            const float ww = w2s[m];
            float h0 = acc0[t][r] + bb; h0 = h0 > 0.f ? h0 : 0.f;
            float h1 = acc1[t][r] + bb; h1 = h1 > 0.f ? h1 : 0.f;
            p0 += h0 * ww;
            p1 += h1 * ww;
        }
    }
    p0 += __shfl_xor(p0, 16, 32);
    p1 += __shfl_xor(p1, 16, 32);
    const float bias2 = b2[0];
    if (lane < 16) {
        out[e0] = p0 + bias2;
        out[e1] = p1 + bias2;
    }
}

// Scalar cleanup for a non-multiple-of-256 edge tail (not taken for E=800000).
__global__ void edge_mlp_tail(const float* __restrict__ node_emb,
                              const long long* __restrict__ edge_index,
                              const float* __restrict__ W1,
                              const float* __restrict__ b1,
                              const float* __restrict__ W2,
                              const float* __restrict__ b2,
                              float* __restrict__ out,
                              int n_edges, int start)
{
    const int e = start + blockIdx.x * blockDim.x + threadIdx.x;
    if (e >= n_edges) return;
    const float* s = node_emb + (size_t)edge_index[e] * HIDDEN;
    const float* d = node_emb + (size_t)edge_index[n_edges + e] * HIDDEN;
    float acc = b2[0];
    for (int m = 0; m < MLP; ++m) {
        float h = b1[m];
        for (int k = 0; k < HIDDEN; ++k)
            h += s[k] * W1[k * MLP + m] + d[k] * W1[(HIDDEN + k) * MLP + m];
        acc += (h > 0.f ? h : 0.f) * W2[m];
    }
    out[e] = acc;
}

extern "C" void kernel_launch(void* const* d_in, const int* in_sizes, int n_in,
                              void* d_out, int out_size, void* d_ws, size_t ws_size,
                              hipStream_t stream) {
    const float*     node_emb   = (const float*)d_in[0];
    const long long* edge_index = (const long long*)d_in[1];
    const float*     W1         = (const float*)d_in[2];
    const float*     b1         = (const float*)d_in[3];
    const float*     W2         = (const float*)d_in[4];
    const float*     b2         = (const float*)d_in[5];
    float*           out        = (float*)d_out;

    const int n_edges = in_sizes[1] / 2;                 // edge_index is [2, E]
    const int nblk    = n_edges / EDGES_PER_BLOCK;       // 3125 for E = 800000
    const size_t lds_bytes = (size_t)(MLP * PITCH + 2 * MLP) * sizeof(float); // ~67KB of 320KB/WGP

    if (nblk > 0) {
        hipLaunchKernelGGL(edge_mlp_wmma, dim3(nblk), dim3(THREADS), lds_bytes, stream,
                           node_emb, edge_index, W1, b1, W2, b2, out, n_edges);
    }
    const int done = nblk * EDGES_PER_BLOCK;
    const int rem  = n_edges - done;
    if (rem > 0) {
        hipLaunchKernelGGL(edge_mlp_tail, dim3((rem + 255) / 256), dim3(256), 0, stream,
                           node_emb, edge_index, W1, b1, W2, b2, out, n_edges, done);
    }
}